// Sparsemax_171798691846
// MI455X (gfx1250) — compile-verified
//
#include <hip/hip_runtime.h>
#include <hip/hip_bf16.h>
#include <stdint.h>

#define N       4096
#define BLOCK   256
#define NWAVES  (BLOCK / 32)
#define VPT     (N / BLOCK)      // 16 elements per thread

typedef __attribute__((ext_vector_type(4))) unsigned int u32x4;
typedef __attribute__((ext_vector_type(8))) int          i32x8;
typedef __attribute__((ext_vector_type(4))) int          i32x4;

__device__ __forceinline__ float wave_sum(float v) {
#pragma unroll
    for (int off = 16; off > 0; off >>= 1) v += __shfl_xor(v, off, 32);
    return v;
}

__device__ __forceinline__ float wave_max(float v) {
#pragma unroll
    for (int off = 16; off > 0; off >>= 1) v = fmaxf(v, __shfl_xor(v, off, 32));
    return v;
}

__global__ __launch_bounds__(BLOCK) void sparsemax_tdm_kernel(
    const float* __restrict__ x, float* __restrict__ out, int rows)
{
    __shared__ float row[N];                 // 16 KB staging for one row
    __shared__ float redbuf[2 * NWAVES];     // cross-wave reduction scratch

    const int tid  = threadIdx.x;
    const int lane = tid & 31;
    const int wave = tid >> 5;
    const int r    = blockIdx.x;

    const uint64_t gbase = (uint64_t)(uintptr_t)(x + (size_t)r * N);

#if __has_builtin(__builtin_amdgcn_tensor_load_to_lds)
    // ---- Tensor Data Mover: DMA the full 16KB row into LDS (wave 0 issues) ----
    if (tid < 32) {
        const unsigned lds_addr = (unsigned)(uintptr_t)(&row[0]);
        // D# group 0 (CDNA5 ISA):
        //  [1:0] count=1 | [63:32] lds_addr | [120:64] global_addr | [127:126] type=2
        u32x4 g0 = { 1u,
                     lds_addr,
                     (unsigned)(gbase & 0xffffffffu),
                     ((unsigned)(gbase >> 32) & 0x01ffffffu) | (2u << 30) };
        // D# group 1 (§8.4): data_size=2 (4B) @ [17:16];
        //  tensor_dim0=4096 @ [79:48]; tensor_dim1=1 @ [111:80];
        //  tile_dim0=4096 @ [127:112]; tile_dim1=tile_dim2=0 (1-D tile);
        //  tensor_dim0_stride=4096 @ [207:160]
        i32x8 g1 = { (int)(2u << 16),
                     (int)((unsigned)(N & 0xffff) << 16),   // tensor_dim0 low16
                     (int)(1u << 16),                        // tensor_dim1 low16
                     (int)((unsigned)(N & 0xffff) << 16),   // tile_dim0
                     0,
                     (int)N,                                 // dim0 stride low32
                     0, 0 };
        i32x4 gz4 = { 0, 0, 0, 0 };                  // groups 2/3 unused (1-D tile)
        i32x8 gz8 = { 0, 0, 0, 0, 0, 0, 0, 0 };      // extra group (this toolchain's 6-arg form)
        __builtin_amdgcn_tensor_load_to_lds(g0, g1, gz4, gz4, gz8, 0);
#if __has_builtin(__builtin_amdgcn_s_wait_tensorcnt)
        __builtin_amdgcn_s_wait_tensorcnt(0);
#else
        asm volatile("s_wait_tensorcnt 0" ::: "memory");
#endif
    }
#else
    // Fallback: cooperative vector loads (should not be taken on this toolchain)
    {
        const float4* src = (const float4*)(x + (size_t)r * N);
        float4* dst = (float4*)row;
#pragma unroll
        for (int j = 0; j < VPT / 4; ++j) dst[tid + j * BLOCK] = src[tid + j * BLOCK];
    }
#endif
    __syncthreads();   // LDS row visible to all 8 waves

    // ---- pull this thread's 16 elements into registers (bank-conflict-free) ----
    float v[VPT];
#pragma unroll
    for (int j = 0; j < VPT; ++j) v[j] = row[tid + j * BLOCK];

    // ---- row max (numerical shift, matches reference) ----
    float m = v[0];
#pragma unroll
    for (int j = 1; j < VPT; ++j) m = fmaxf(m, v[j]);
    m = wave_max(m);
    __syncthreads();
    if (lane == 0) redbuf[wave] = m;
    __syncthreads();
    m = redbuf[0];
#pragma unroll
    for (int i = 1; i < NWAVES; ++i) m = fmaxf(m, redbuf[i]);

    // ---- initial tau over full support: (sum(y) - 1)/n ----
    float ps = 0.0f;
#pragma unroll
    for (int j = 0; j < VPT; ++j) ps += v[j] - m;
    ps = wave_sum(ps);
    __syncthreads();
    if (lane == 0) redbuf[wave] = ps;
    __syncthreads();
    float S = 0.0f;
#pragma unroll
    for (int i = 0; i < NWAVES; ++i) S += redbuf[i];
    float tau = (S - 1.0f) / (float)N;

    // ---- Michelot fixed point: support shrinks, tau rises, exact on convergence ----
    for (int it = 0; it < 32; ++it) {
        float s = 0.0f, c = 0.0f;
#pragma unroll
        for (int j = 0; j < VPT; ++j) {
            float y = v[j] - m;
            if (y > tau) { s += y; c += 1.0f; }
        }
        s = wave_sum(s);
        c = wave_sum(c);
        __syncthreads();
        if (lane == 0) { redbuf[wave] = s; redbuf[NWAVES + wave] = c; }
        __syncthreads();
        float Ssum = 0.0f, Csum = 0.0f;
#pragma unroll
        for (int i = 0; i < NWAVES; ++i) { Ssum += redbuf[i]; Csum += redbuf[NWAVES + i]; }
        float tn = (Ssum - 1.0f) / Csum;   // identical on every thread -> uniform branch
        if (tn == tau) break;
        tau = tn;
    }

    // ---- coalesced output: max(0, y - tau) ----
    const size_t ob = (size_t)r * N;
#pragma unroll
    for (int j = 0; j < VPT; ++j)
        out[ob + tid + j * BLOCK] = fmaxf(0.0f, (v[j] - m) - tau);
}

extern "C" void kernel_launch(void* const* d_in, const int* in_sizes, int n_in,
                              void* d_out, int out_size, void* d_ws, size_t ws_size,
                              hipStream_t stream)
{
    const float* x  = (const float*)d_in[0];
    float* out      = (float*)d_out;
    const int rows  = in_sizes[0] / N;   // 8192
    sparsemax_tdm_kernel<<<rows, BLOCK, 0, stream>>>(x, out, rows);
}